// EGNN_44006234915569
// MI455X (gfx1250) — compile-verified
//
#include <hip/hip_runtime.h>
#include <hip/hip_bf16.h>

#define N_NODES 50000
#define N_EDGES 400000
#define HID 128
#define LSTR 132   // LDS row stride in floats (conflict-free: 132 % 64 = 4)
#define WAVES 4    // waves per block for WMMA kernels

typedef float v2f __attribute__((ext_vector_type(2)));
typedef float v8f __attribute__((ext_vector_type(8)));

#define DSWAIT() asm volatile("s_wait_dscnt 0" ::: "memory")

// D(16x16,f32) += A(16x4,f32) * B(4x16,f32)  -- CDNA5 fp32 WMMA
__device__ __forceinline__ v8f wmma4v(v2f a, v2f b, v8f c) {
    return __builtin_amdgcn_wmma_f32_16x16x4_f32(false, a, false, b, (short)0, c,
                                                 false, false);
}
__device__ __forceinline__ v2f ld2(const float* p) { return *(const v2f*)p; }

// ---------------------------------------------------------------------------
// Weight transpose: WT[n*K + k] = W[k*N + n]   (run once per launch; tiny)
// ---------------------------------------------------------------------------
__global__ void transpose_kernel(const float* __restrict__ in,
                                 float* __restrict__ outp, int K, int N)
{
    int idx = blockIdx.x * blockDim.x + threadIdx.x;
    if (idx >= K * N) return;
    int k = idx / N, n = idx - k * N;
    outp[(size_t)n * K + k] = in[idx];
}

// ---------------------------------------------------------------------------
// Embed: h = h0[n,16] @ W[16,128] + B     (WT layout [128][16])
// ---------------------------------------------------------------------------
__global__ __launch_bounds__(WAVES * 32) void embed_kernel(
    const float* __restrict__ h0, const float* __restrict__ WT,
    const float* __restrict__ B, float* __restrict__ h, int n)
{
    const int lane = threadIdx.x & 31;
    const int wave = threadIdx.x >> 5;
    const int m    = lane & 15;
    const int hk   = lane >> 4;
    const int tile = blockIdx.x * WAVES + wave;
    const int i0   = tile * 16;
    if (i0 >= n) return;

    const float* arow = h0 + (size_t)(i0 + m) * 16;
    v2f a[4];
#pragma unroll
    for (int t = 0; t < 4; ++t) a[t] = ld2(arow + 4 * t + 2 * hk);

    for (int nt = 0; nt < 8; ++nt) {
        int cn = nt * 16 + m;
        const float* wr = WT + (size_t)cn * 16;
        v8f acc = {0, 0, 0, 0, 0, 0, 0, 0};
#pragma unroll
        for (int t = 0; t < 4; ++t)
            acc = wmma4v(a[t], ld2(wr + 4 * t + 2 * hk), acc);
        float bias = B[cn];
#pragma unroll
        for (int v = 0; v < 8; ++v)
            h[(size_t)(i0 + v + 8 * hk) * HID + cn] = acc[v] + bias;
    }
}

// ---------------------------------------------------------------------------
// Geometry: rel_dist, rel_pos_spin2, segment counts
// ---------------------------------------------------------------------------
__global__ void geom_kernel(const float* __restrict__ x,
                            const int* __restrict__ row_idx,
                            const int* __restrict__ col_idx,
                            float* __restrict__ rel_dist,
                            float* __restrict__ spin2,
                            float* __restrict__ cnt, int n_edges)
{
    int e = blockIdx.x * blockDim.x + threadIdx.x;
    if (e >= n_edges) return;
    int r = row_idx[e], c = col_idx[e];
    float rx = x[(size_t)r * 3 + 0] - x[(size_t)c * 3 + 0];
    float ry = x[(size_t)r * 3 + 1] - x[(size_t)c * 3 + 1];
    float rz = x[(size_t)r * 3 + 2] - x[(size_t)c * 3 + 2];
    float rxy2 = rx * rx + ry * ry;
    float d  = sqrtf(rxy2 + rz * rz);
    float th = atan2f(ry, rx);
    float ph = atan2f(rz, sqrtf(rxy2));
    rel_dist[e] = d;
    spin2[(size_t)e * 3 + 0] = d * cosf(2.0f * th);
    spin2[(size_t)e * 3 + 1] = d * sinf(2.0f * th);
    spin2[(size_t)e * 3 + 2] = d * ph;
    atomicAdd(&cnt[r], 1.0f);
}

// ---------------------------------------------------------------------------
// Edge layer (layers 1..3): ef = MLP_e([h[row],h[col],dist]); agg[row] += ef
// WT1: [128][256] transposed (cols 0..127 = h[row] part, 128..255 = h[col]);
// W1row256: original W1 (for dist row 256). WT2: [128][128] transposed.
// ---------------------------------------------------------------------------
__global__ __launch_bounds__(WAVES * 32) void edge_layer_kernel(
    const float* __restrict__ h,
    const int* __restrict__ row_idx, const int* __restrict__ col_idx,
    const float* __restrict__ rel_dist,
    const float* __restrict__ WT1, const float* __restrict__ W1,
    const float* __restrict__ B1,
    const float* __restrict__ WT2, const float* __restrict__ B2,
    float* __restrict__ agg, int n_edges)
{
    __shared__ float lds[WAVES][16 * LSTR];
    const int lane = threadIdx.x & 31;
    const int wave = threadIdx.x >> 5;
    const int m    = lane & 15;
    const int hk   = lane >> 4;
    const int tile = blockIdx.x * WAVES + wave;
    const int e0   = tile * 16;
    if (e0 >= n_edges) return;
    float* sh = lds[wave];

    int er = row_idx[e0 + m], ec = col_idx[e0 + m];
    const float* hr = h + (size_t)er * HID;
    const float* hc = h + (size_t)ec * HID;
    v2f aR[32], aC[32];
#pragma unroll
    for (int t = 0; t < 32; ++t) {
        int k = 4 * t + 2 * hk;
        aR[t] = ld2(hr + k);
        aC[t] = ld2(hc + k);
    }
    float rd[8];
    int   rarr[8];
#pragma unroll
    for (int v = 0; v < 8; ++v) {
        int e = e0 + v + 8 * hk;
        rd[v] = rel_dist[e];
        rarr[v] = row_idx[e];
    }

    // GEMM1: hidden = relu([hr|hc|dist] @ W1 + B1)  -> LDS (row-major tile)
    for (int nt = 0; nt < 8; ++nt) {
        int cn = nt * 16 + m;
        const float* w1r = WT1 + (size_t)cn * 256;
        v8f acc = {0, 0, 0, 0, 0, 0, 0, 0};
#pragma unroll
        for (int t = 0; t < 32; ++t) {
            int k = 4 * t + 2 * hk;
            acc = wmma4v(aR[t], ld2(w1r + k), acc);
            acc = wmma4v(aC[t], ld2(w1r + 128 + k), acc);
        }
        float bias = B1[cn];
        float w1c  = W1[(size_t)256 * HID + cn];
#pragma unroll
        for (int v = 0; v < 8; ++v)
            sh[(v + 8 * hk) * LSTR + cn] = fmaxf(acc[v] + bias + rd[v] * w1c, 0.0f);
    }
    DSWAIT();

    // GEMM2: ef = hidden @ W2 + B2, atomic into agg[row]
    for (int nt = 0; nt < 8; ++nt) {
        int cn = nt * 16 + m;
        const float* w2r = WT2 + (size_t)cn * 128;
        v8f acc = {0, 0, 0, 0, 0, 0, 0, 0};
#pragma unroll
        for (int t = 0; t < 32; ++t) {
            int k = 4 * t + 2 * hk;
            acc = wmma4v(ld2(sh + m * LSTR + k), ld2(w2r + k), acc);
        }
        float bias = B2[cn];
#pragma unroll
        for (int v = 0; v < 8; ++v)
            atomicAdd(&agg[(size_t)rarr[v] * HID + cn], acc[v] + bias);
    }
}

// ---------------------------------------------------------------------------
// Node layer: h = h + MLP_n([h, agg])   (WT1: [128][256], WT2: [128][128])
// ---------------------------------------------------------------------------
__global__ __launch_bounds__(WAVES * 32) void node_layer_kernel(
    float* __restrict__ h, const float* __restrict__ agg,
    const float* __restrict__ WT1, const float* __restrict__ B1,
    const float* __restrict__ WT2, const float* __restrict__ B2, int n)
{
    __shared__ float lds[WAVES][16 * LSTR];
    const int lane = threadIdx.x & 31;
    const int wave = threadIdx.x >> 5;
    const int m    = lane & 15;
    const int hk   = lane >> 4;
    const int tile = blockIdx.x * WAVES + wave;
    const int i0   = tile * 16;
    if (i0 >= n) return;
    float* sh = lds[wave];

    const float* hR = h   + (size_t)(i0 + m) * HID;
    const float* hA = agg + (size_t)(i0 + m) * HID;
    v2f aH[32], aA[32];
#pragma unroll
    for (int t = 0; t < 32; ++t) {
        int k = 4 * t + 2 * hk;
        aH[t] = ld2(hR + k);
        aA[t] = ld2(hA + k);
    }

    for (int nt = 0; nt < 8; ++nt) {
        int cn = nt * 16 + m;
        const float* w1r = WT1 + (size_t)cn * 256;
        v8f acc = {0, 0, 0, 0, 0, 0, 0, 0};
#pragma unroll
        for (int t = 0; t < 32; ++t) {
            int k = 4 * t + 2 * hk;
            acc = wmma4v(aH[t], ld2(w1r + k), acc);
            acc = wmma4v(aA[t], ld2(w1r + 128 + k), acc);
        }
        float bias = B1[cn];
#pragma unroll
        for (int v = 0; v < 8; ++v)
            sh[(v + 8 * hk) * LSTR + cn] = fmaxf(acc[v] + bias, 0.0f);
    }
    DSWAIT();

    for (int nt = 0; nt < 8; ++nt) {
        int cn = nt * 16 + m;
        const float* w2r = WT2 + (size_t)cn * 128;
        v8f acc = {0, 0, 0, 0, 0, 0, 0, 0};
#pragma unroll
        for (int t = 0; t < 32; ++t) {
            int k = 4 * t + 2 * hk;
            acc = wmma4v(ld2(sh + m * LSTR + k), ld2(w2r + k), acc);
        }
        float bias = B2[cn];
#pragma unroll
        for (int v = 0; v < 8; ++v) {
            size_t o = (size_t)(i0 + v + 8 * hk) * HID + cn;
            h[o] = h[o] + acc[v] + bias;
        }
    }
}

// ---------------------------------------------------------------------------
// Layer 4 fused: ef = edgeMLP(...), w = weightMLP(ef),
//                v_latent[row] += w * spin2   (no node update needed)
// ---------------------------------------------------------------------------
__global__ __launch_bounds__(WAVES * 32) void edge_final_kernel(
    const float* __restrict__ h,
    const int* __restrict__ row_idx, const int* __restrict__ col_idx,
    const float* __restrict__ rel_dist, const float* __restrict__ spin2,
    const float* __restrict__ WeT1, const float* __restrict__ We1,
    const float* __restrict__ Be1,
    const float* __restrict__ WeT2, const float* __restrict__ Be2,
    const float* __restrict__ WgT1, const float* __restrict__ Bg1,
    const float* __restrict__ WgT2, const float* __restrict__ Bg2,
    float* __restrict__ vlat, int n_edges)
{
    __shared__ float lds[WAVES][16 * LSTR];
    const int lane = threadIdx.x & 31;
    const int wave = threadIdx.x >> 5;
    const int m    = lane & 15;
    const int hk   = lane >> 4;
    const int tile = blockIdx.x * WAVES + wave;
    const int e0   = tile * 16;
    if (e0 >= n_edges) return;
    float* sh = lds[wave];

    int er = row_idx[e0 + m], ec = col_idx[e0 + m];
    const float* hr = h + (size_t)er * HID;
    const float* hc = h + (size_t)ec * HID;
    v2f aR[32], aC[32];
#pragma unroll
    for (int t = 0; t < 32; ++t) {
        int k = 4 * t + 2 * hk;
        aR[t] = ld2(hr + k);
        aC[t] = ld2(hc + k);
    }
    float rd[8];
    int   rarr[8];
    float s0a[8], s1a[8], s2a[8];
#pragma unroll
    for (int v = 0; v < 8; ++v) {
        int e = e0 + v + 8 * hk;
        rd[v]   = rel_dist[e];
        rarr[v] = row_idx[e];
        s0a[v]  = spin2[(size_t)e * 3 + 0];
        s1a[v]  = spin2[(size_t)e * 3 + 1];
        s2a[v]  = spin2[(size_t)e * 3 + 2];
    }

    // Phase 1: hidden1 = relu(edge inputs @ We1 + Be1) -> LDS
    for (int nt = 0; nt < 8; ++nt) {
        int cn = nt * 16 + m;
        const float* w1r = WeT1 + (size_t)cn * 256;
        v8f acc = {0, 0, 0, 0, 0, 0, 0, 0};
#pragma unroll
        for (int t = 0; t < 32; ++t) {
            int k = 4 * t + 2 * hk;
            acc = wmma4v(aR[t], ld2(w1r + k), acc);
            acc = wmma4v(aC[t], ld2(w1r + 128 + k), acc);
        }
        float bias = Be1[cn];
        float w1c  = We1[(size_t)256 * HID + cn];
#pragma unroll
        for (int v = 0; v < 8; ++v)
            sh[(v + 8 * hk) * LSTR + cn] = fmaxf(acc[v] + bias + rd[v] * w1c, 0.0f);
    }
    DSWAIT();

    // Phase 2: ef = hidden1 @ We2 + Be2  (regs, then LDS)
    v8f ef[8];
    for (int nt = 0; nt < 8; ++nt) {
        int cn = nt * 16 + m;
        const float* w2r = WeT2 + (size_t)cn * 128;
        v8f acc = {0, 0, 0, 0, 0, 0, 0, 0};
#pragma unroll
        for (int t = 0; t < 32; ++t) {
            int k = 4 * t + 2 * hk;
            acc = wmma4v(ld2(sh + m * LSTR + k), ld2(w2r + k), acc);
        }
        float bias = Be2[cn];
#pragma unroll
        for (int v = 0; v < 8; ++v) acc[v] += bias;
        ef[nt] = acc;
    }
    DSWAIT();
    for (int nt = 0; nt < 8; ++nt) {
        int cn = nt * 16 + m;
#pragma unroll
        for (int v = 0; v < 8; ++v) sh[(v + 8 * hk) * LSTR + cn] = ef[nt][v];
    }
    DSWAIT();

    // Phase 3: hidden2 = relu(ef @ Wg1 + Bg1)  (regs, then LDS)
    v8f h2[8];
    for (int nt = 0; nt < 8; ++nt) {
        int cn = nt * 16 + m;
        const float* w3r = WgT1 + (size_t)cn * 128;
        v8f acc = {0, 0, 0, 0, 0, 0, 0, 0};
#pragma unroll
        for (int t = 0; t < 32; ++t) {
            int k = 4 * t + 2 * hk;
            acc = wmma4v(ld2(sh + m * LSTR + k), ld2(w3r + k), acc);
        }
        float bias = Bg1[cn];
#pragma unroll
        for (int v = 0; v < 8; ++v) acc[v] = fmaxf(acc[v] + bias, 0.0f);
        h2[nt] = acc;
    }
    DSWAIT();
    for (int nt = 0; nt < 8; ++nt) {
        int cn = nt * 16 + m;
#pragma unroll
        for (int v = 0; v < 8; ++v) sh[(v + 8 * hk) * LSTR + cn] = h2[nt][v];
    }
    DSWAIT();

    // Phase 4: w = hidden2 @ Wg2 + Bg2; scatter w * spin2 into v_latent
    for (int nt = 0; nt < 8; ++nt) {
        int cn = nt * 16 + m;
        const float* w4r = WgT2 + (size_t)cn * 128;
        v8f acc = {0, 0, 0, 0, 0, 0, 0, 0};
#pragma unroll
        for (int t = 0; t < 32; ++t) {
            int k = 4 * t + 2 * hk;
            acc = wmma4v(ld2(sh + m * LSTR + k), ld2(w4r + k), acc);
        }
        float bias = Bg2[cn];
#pragma unroll
        for (int v = 0; v < 8; ++v) {
            float wv = acc[v] + bias;
            size_t base = ((size_t)rarr[v] * HID + cn) * 3;
            atomicAdd(&vlat[base + 0], wv * s0a[v]);
            atomicAdd(&vlat[base + 1], wv * s1a[v]);
            atomicAdd(&vlat[base + 2], wv * s2a[v]);
        }
    }
}

// ---------------------------------------------------------------------------
// Finalize: v_latent /= max(cnt,1); copy x; v = normalize(v_latent[:,0,:2])
// ---------------------------------------------------------------------------
__global__ void finalize_kernel(float* __restrict__ out,
                                const float* __restrict__ x,
                                const float* __restrict__ cnt, int n)
{
    int idx = blockIdx.x * blockDim.x + threadIdx.x;
    if (idx >= n * HID) return;
    int i = idx >> 7, c = idx & (HID - 1);
    float inv = 1.0f / fmaxf(cnt[i], 1.0f);
    float* vl = out + (size_t)idx * 3;
    float v0 = vl[0] * inv, v1 = vl[1] * inv, v2 = vl[2] * inv;
    vl[0] = v0; vl[1] = v1; vl[2] = v2;
    float* xout = out + (size_t)n * HID * 3;
    float* vout = xout + (size_t)n * 3;
    if (c == 1) {
        xout[(size_t)i * 3 + 0] = x[(size_t)i * 3 + 0];
        xout[(size_t)i * 3 + 1] = x[(size_t)i * 3 + 1];
        xout[(size_t)i * 3 + 2] = x[(size_t)i * 3 + 2];
    }
    if (c == 0) {
        float nrm = fmaxf(sqrtf(v0 * v0 + v1 * v1), 1e-12f);
        vout[(size_t)i * 2 + 0] = v0 / nrm;
        vout[(size_t)i * 2 + 1] = v1 / nrm;
    }
}

// ---------------------------------------------------------------------------
static inline size_t align256(size_t x) { return (x + 255) & ~(size_t)255; }

extern "C" void kernel_launch(void* const* d_in, const int* in_sizes, int n_in,
                              void* d_out, int out_size, void* d_ws, size_t ws_size,
                              hipStream_t stream)
{
    const float* h0   = (const float*)d_in[0];   // [50000,16]
    const float* x    = (const float*)d_in[1];   // [50000,3]
    const int*   eidx = (const int*)d_in[2];     // [2,400000]
    const float* Wemb = (const float*)d_in[3];
    const float* Bemb = (const float*)d_in[4];
    const float* We1  = (const float*)d_in[5];
    const float* Be1  = (const float*)d_in[6];
    const float* We2  = (const float*)d_in[7];
    const float* Be2  = (const float*)d_in[8];
    const float* Wn1  = (const float*)d_in[9];
    const float* Bn1  = (const float*)d_in[10];
    const float* Wn2  = (const float*)d_in[11];
    const float* Bn2  = (const float*)d_in[12];
    const float* Wg1  = (const float*)d_in[13];
    const float* Bg1  = (const float*)d_in[14];
    const float* Wg2  = (const float*)d_in[15];
    const float* Bg2  = (const float*)d_in[16];

    const int* erow = eidx;
    const int* ecol = eidx + N_EDGES;
    float* out = (float*)d_out;

    char* p = (char*)d_ws;
    float* h    = (float*)p; p += align256((size_t)N_NODES * HID * sizeof(float));
    float* agg  = (float*)p; p += align256((size_t)N_NODES * HID * sizeof(float));
    float* rdst = (float*)p; p += align256((size_t)N_EDGES * sizeof(float));
    float* sp2  = (float*)p; p += align256((size_t)N_EDGES * 3 * sizeof(float));
    float* cnt  = (float*)p; p += align256((size_t)N_NODES * sizeof(float));
    float* WembT= (float*)p; p += align256((size_t)16  * HID * sizeof(float));
    float* WeT1 = (float*)p; p += align256((size_t)256 * HID * sizeof(float));
    float* WeT2 = (float*)p; p += align256((size_t)HID * HID * sizeof(float));
    float* WnT1 = (float*)p; p += align256((size_t)256 * HID * sizeof(float));
    float* WnT2 = (float*)p; p += align256((size_t)HID * HID * sizeof(float));
    float* WgT1 = (float*)p; p += align256((size_t)HID * HID * sizeof(float));
    float* WgT2 = (float*)p; p += align256((size_t)HID * HID * sizeof(float));

    const int node_tiles = N_NODES / 16;                        // 3125
    const int edge_tiles = N_EDGES / 16;                        // 25000
    const int node_blocks = (node_tiles + WAVES - 1) / WAVES;   // 782
    const int edge_blocks = (edge_tiles + WAVES - 1) / WAVES;   // 6250
    const dim3 blk(WAVES * 32);

    hipMemsetAsync(out, 0, (size_t)N_NODES * HID * 3 * sizeof(float), stream);
    hipMemsetAsync(cnt, 0, (size_t)N_NODES * sizeof(float), stream);

    // One-time weight transposes (B fragments become contiguous b64 loads)
    transpose_kernel<<<(16 * HID + 255) / 256, 256, 0, stream>>>(Wemb, WembT, 16, HID);
    transpose_kernel<<<(256 * HID + 255) / 256, 256, 0, stream>>>(We1, WeT1, 256, HID);
    transpose_kernel<<<(HID * HID + 255) / 256, 256, 0, stream>>>(We2, WeT2, HID, HID);
    transpose_kernel<<<(256 * HID + 255) / 256, 256, 0, stream>>>(Wn1, WnT1, 256, HID);
    transpose_kernel<<<(HID * HID + 255) / 256, 256, 0, stream>>>(Wn2, WnT2, HID, HID);
    transpose_kernel<<<(HID * HID + 255) / 256, 256, 0, stream>>>(Wg1, WgT1, HID, HID);
    transpose_kernel<<<(HID * HID + 255) / 256, 256, 0, stream>>>(Wg2, WgT2, HID, HID);

    embed_kernel<<<node_blocks, blk, 0, stream>>>(h0, WembT, Bemb, h, N_NODES);
    geom_kernel<<<(N_EDGES + 255) / 256, 256, 0, stream>>>(x, erow, ecol,
                                                           rdst, sp2, cnt, N_EDGES);

    for (int layer = 0; layer < 3; ++layer) {
        hipMemsetAsync(agg, 0, (size_t)N_NODES * HID * sizeof(float), stream);
        edge_layer_kernel<<<edge_blocks, blk, 0, stream>>>(
            h, erow, ecol, rdst, WeT1, We1, Be1, WeT2, Be2, agg, N_EDGES);
        node_layer_kernel<<<node_blocks, blk, 0, stream>>>(
            h, agg, WnT1, Bn1, WnT2, Bn2, N_NODES);
    }

    edge_final_kernel<<<edge_blocks, blk, 0, stream>>>(
        h, erow, ecol, rdst, sp2, WeT1, We1, Be1, WeT2, Be2,
        WgT1, Bg1, WgT2, Bg2, out, N_EDGES);

    finalize_kernel<<<(N_NODES * HID + 255) / 256, 256, 0, stream>>>(
        out, x, cnt, N_NODES);
}